// loss_generator_11785390260824
// MI455X (gfx1250) — compile-verified
//
#include <hip/hip_runtime.h>

typedef __attribute__((ext_vector_type(2))) float v2f;
typedef __attribute__((ext_vector_type(8))) float v8f;

#define T_DIM     512
#define HWDIM     257
#define HWSQ      (HWDIM * HWDIM)     // 66049
#define U_TSTRIDE (2 * HWSQ)          // 132098  (output has C=2, u = channel 0)
#define INT_W     255
#define NPIX      (INT_W * INT_W)     // 65025 interior pixels
#define NTILE     32                  // pixels per workgroup (two 16-wide WMMA N tiles)
#define NPXT      ((NPIX + NTILE - 1) / NTILE)  // 2033 pixel tiles
#define KSTEPS    (T_DIM / 4)         // 128 WMMA K-steps (K=4 per wmma)
#define KCHUNK    256                 // timesteps staged in LDS per phase
#define KSTEPS_PH (KCHUNK / 4)        // 64 K-steps per phase

#define SCALE_C   3.5682482323055424f     // dt^(-0.5)/Gamma(1.5)
#define KAPPA_C   0.01f
#define INV_DX2   16384.0f                // 1/DX^2, DX = 1/128
#define W_CRN     0.34520446044393f
#define W_EDG     0.309591078922457f
#define W_CTR     (-2.619182157203629f)
#define INV_N     (1.0f / 33292800.0f)    // 1/(512*255*255)

// M[i,k] for ALPHA=0.5:  w[j] = sqrt(j+1)-sqrt(j);  dvec[d] = 2*sqrt(d)-sqrt(d-1)-sqrt(d+1)
__device__ __forceinline__ float m_entry(int i, int k) {
    if (i == 0) return 0.0f;
    if (k == i) return 1.0f;
    if (k == 0) {
        float j = (float)(i - 1);
        return -(sqrtf(j + 1.0f) - sqrtf(j));
    }
    if (k >= 1 && k < i) {
        float d = (float)(i - k);  // 1..510
        return -(2.0f * sqrtf(d) - sqrtf(d - 1.0f) - sqrtf(d + 1.0f));
    }
    return 0.0f;
}

// Pack scale*M into WMMA A-fragment order:
// index = ((itile*128 + kk)*32 + lane), lane<16 -> row itile*16+lane, K = 4kk+{0,1}
//                                       lane>=16 -> row itile*16+lane-16, K = 4kk+{2,3}
__global__ __launch_bounds__(256) void pack_M_kernel(v2f* __restrict__ Apack) {
    int idx  = blockIdx.x * 256 + threadIdx.x;   // [0, 32*128*32) = 131072
    int lane = idx & 31;
    int kk   = (idx >> 5) & 127;
    int it   = idx >> 12;
    int i    = it * 16 + (lane & 15);
    int kb   = kk * 4 + ((lane >> 4) << 1);
    v2f a;
    a.x = SCALE_C * m_entry(i, kb);
    a.y = SCALE_C * m_entry(i, kb + 1);
    Apack[idx] = a;
}

#define LDS_PANEL (KCHUNK * 16)      // 4096 floats per pixel-half panel

__global__ __launch_bounds__(256) void frac_loss_kernel(const float* __restrict__ outp,
                                                        const float* __restrict__ f1p,
                                                        const v2f*  __restrict__ Apack,
                                                        float*      __restrict__ partial) {
    // Two 256x16 B panels; +16 pad so panel1 banks are offset by 16 -> conflict-free
    __shared__ float ldsU[2 * LDS_PANEL + 16];
    __shared__ float red[8];
    float* U0 = ldsU;
    float* U1 = ldsU + LDS_PANEL + 16;

    const int tid     = threadIdx.x;
    const int px_base = blockIdx.x * NTILE;

    const int lane = tid & 31;
    const int wave = tid >> 5;
    const int hl   = lane >> 4;    // half-wave select (K pair / row+8)
    const int j    = lane & 15;    // column within 16-wide N tile

    // A-fragment base pointers for this wave's 4 i-tiles
    const v2f* A0 = Apack + (size_t)((wave * 4 + 0) * KSTEPS) * 32 + lane;
    const v2f* A1 = Apack + (size_t)((wave * 4 + 1) * KSTEPS) * 32 + lane;
    const v2f* A2 = Apack + (size_t)((wave * 4 + 2) * KSTEPS) * 32 + lane;
    const v2f* A3 = Apack + (size_t)((wave * 4 + 3) * KSTEPS) * 32 + lane;

    v8f acc00 = {}, acc01 = {};   // i-tile 0 x {pixels 0-15, pixels 16-31}
    v8f acc10 = {}, acc11 = {};
    v8f acc20 = {}, acc21 = {};
    v8f acc30 = {}, acc31 = {};

    for (int ph = 0; ph < 2; ++ph) {
        // ---- Stage B panels: u columns for 32 interior pixels, 256 timesteps ----
        for (int e = tid; e < KCHUNK * NTILE; e += 256) {
            int kloc = e >> 5;              // 0..255
            int jj   = e & 31;              // pixel within WG tile
            int px   = px_base + jj;
            float v  = 0.0f;
            if (px < NPIX) {
                int h = 1 + px / INT_W;
                int w = 1 + px % INT_W;
                v = outp[(size_t)(ph * KCHUNK + kloc) * U_TSTRIDE + h * HWDIM + w];
            }
            float* dst = (jj & 16) ? U1 : U0;
            dst[kloc * 16 + (jj & 15)] = v;
        }
        __syncthreads();

        // ---- GEMM phase: 64 K-steps, 8 WMMAs per step, A reused across 2 N tiles ----
#pragma unroll 2
        for (int kk2 = 0; kk2 < KSTEPS_PH; ++kk2) {
            const int kk = ph * KSTEPS_PH + kk2;    // global K-step for A
            const int kb = kk2 * 4 + hl * 2;        // local K row pair in panel
            v2f b0, b1;
            b0.x = U0[kb * 16 + j];
            b0.y = U0[(kb + 1) * 16 + j];
            b1.x = U1[kb * 16 + j];
            b1.y = U1[(kb + 1) * 16 + j];
            v2f a0 = A0[kk * 32];
            v2f a1 = A1[kk * 32];
            v2f a2 = A2[kk * 32];
            v2f a3 = A3[kk * 32];
            acc00 = __builtin_amdgcn_wmma_f32_16x16x4_f32(false, a0, false, b0, (short)0, acc00, false, false);
            acc01 = __builtin_amdgcn_wmma_f32_16x16x4_f32(false, a0, false, b1, (short)0, acc01, false, false);
            acc10 = __builtin_amdgcn_wmma_f32_16x16x4_f32(false, a1, false, b0, (short)0, acc10, false, false);
            acc11 = __builtin_amdgcn_wmma_f32_16x16x4_f32(false, a1, false, b1, (short)0, acc11, false, false);
            acc20 = __builtin_amdgcn_wmma_f32_16x16x4_f32(false, a2, false, b0, (short)0, acc20, false, false);
            acc21 = __builtin_amdgcn_wmma_f32_16x16x4_f32(false, a2, false, b1, (short)0, acc21, false, false);
            acc30 = __builtin_amdgcn_wmma_f32_16x16x4_f32(false, a3, false, b0, (short)0, acc30, false, false);
            acc31 = __builtin_amdgcn_wmma_f32_16x16x4_f32(false, a3, false, b1, (short)0, acc31, false, false);
        }
        __syncthreads();   // panels fully consumed before restage
    }

    // ---- Fused residual: res = D - kappa*lap(u)/dx^2 + u - f1 ; accumulate res^2 ----
    // C/D layout: VGPR r, lanes 0-15 -> row i0+r, lanes 16-31 -> row i0+r+8; N = lane%16
    float lsum = 0.0f;
    const int  pxA = px_base + j;          // N tile 0
    const int  pxB = px_base + 16 + j;     // N tile 1
    const bool vA  = pxA < NPIX;
    const bool vB  = pxB < NPIX;
    int hA = 1, wA = 1, hB = 1, wB = 1;
    if (vA) { hA = 1 + pxA / INT_W; wA = 1 + pxA % INT_W; }
    if (vB) { hB = 1 + pxB / INT_W; wB = 1 + pxB % INT_W; }
    const float* ubaseA = outp + hA * HWDIM + wA;
    const float* fbaseA = f1p  + hA * HWDIM + wA;
    const float* ubaseB = outp + hB * HWDIM + wB;
    const float* fbaseB = f1p  + hB * HWDIM + wB;

#define RESID(ACC, ITEXPR, UB, FB, V)                                             \
    {                                                                             \
        const int i0 = (ITEXPR) * 16 + hl * 8;                                    \
        _Pragma("unroll")                                                         \
        for (int r = 0; r < 8; ++r) {                                             \
            if (V) {                                                              \
                const float* up = (UB) + (size_t)(i0 + r) * U_TSTRIDE;            \
                float c   = up[0];                                                \
                float lap = W_CRN * (up[-HWDIM - 1] + up[-HWDIM + 1] +            \
                                     up[ HWDIM - 1] + up[ HWDIM + 1]) +           \
                            W_EDG * (up[-HWDIM] + up[-1] + up[1] + up[HWDIM]) +   \
                            W_CTR * c;                                            \
                float f   = (FB)[(size_t)(i0 + r) * HWSQ];                        \
                float res = ACC[r] - KAPPA_C * (lap * INV_DX2) + c - f;           \
                lsum += res * res;                                                \
            }                                                                     \
        }                                                                         \
    }

    RESID(acc00, wave * 4 + 0, ubaseA, fbaseA, vA)
    RESID(acc01, wave * 4 + 0, ubaseB, fbaseB, vB)
    RESID(acc10, wave * 4 + 1, ubaseA, fbaseA, vA)
    RESID(acc11, wave * 4 + 1, ubaseB, fbaseB, vB)
    RESID(acc20, wave * 4 + 2, ubaseA, fbaseA, vA)
    RESID(acc21, wave * 4 + 2, ubaseB, fbaseB, vB)
    RESID(acc30, wave * 4 + 3, ubaseA, fbaseA, vA)
    RESID(acc31, wave * 4 + 3, ubaseB, fbaseB, vB)
#undef RESID

    // ---- Deterministic reduction: wave shuffle -> LDS -> per-block partial ----
#pragma unroll
    for (int off = 16; off > 0; off >>= 1)
        lsum += __shfl_xor(lsum, off, 32);

    __syncthreads();
    if (lane == 0) red[wave] = lsum;
    __syncthreads();
    if (tid == 0) {
        float t = 0.0f;
#pragma unroll
        for (int q = 0; q < 8; ++q) t += red[q];
        partial[blockIdx.x] = t;
    }
}

__global__ __launch_bounds__(256) void final_reduce_kernel(const float* __restrict__ partial,
                                                           int n, float* __restrict__ out) {
    __shared__ float sh[256];
    float s = 0.0f;
    for (int idx = threadIdx.x; idx < n; idx += 256) s += partial[idx];
    sh[threadIdx.x] = s;
    __syncthreads();
    for (int off = 128; off > 0; off >>= 1) {
        if (threadIdx.x < off) sh[threadIdx.x] += sh[threadIdx.x + off];
        __syncthreads();
    }
    if (threadIdx.x == 0) out[0] = sh[0] * INV_N;
}

extern "C" void kernel_launch(void* const* d_in, const int* in_sizes, int n_in,
                              void* d_out, int out_size, void* d_ws, size_t ws_size,
                              hipStream_t stream) {
    const float* outp = (const float*)d_in[0];   // (512, 2, 257, 257) f32
    const float* f1p  = (const float*)d_in[1];   // (512, 1, 257, 257) f32
    // d_in[2] = lap_kernel (compile-time constants in the reference; hardcoded)

    v2f*   Apack   = (v2f*)d_ws;                                        // 1 MB
    float* partial = (float*)((char*)d_ws + (size_t)T_DIM * T_DIM * 4); // 2033 floats
    float* outv    = (float*)d_out;

    // 1) Build scale*M in WMMA A-fragment layout (32 i-tiles * 128 ksteps * 32 lanes)
    pack_M_kernel<<<(32 * KSTEPS * 32) / 256, 256, 0, stream>>>(Apack);

    // 2) Fused GEMM (v_wmma_f32_16x16x4_f32) + Laplacian + residual, per-block partials
    frac_loss_kernel<<<NPXT, 256, 0, stream>>>(outp, f1p, Apack, partial);

    // 3) Deterministic final reduce + mean
    final_reduce_kernel<<<1, 256, 0, stream>>>(partial, NPXT, outv);
}